// UnlabeledEntity_21337397527385
// MI455X (gfx1250) — compile-verified
//
#include <hip/hip_runtime.h>
#include <hip/hip_bf16.h>
#include <math.h>

typedef __attribute__((ext_vector_type(2))) float v2f;
typedef __attribute__((ext_vector_type(8))) float v8f;

#define BB 2
#define LL 256
#define HH 256
#define NC 5
#define VOCAB 255   // 2*127+1 distinct clipped distances
#define D4H 1024    // 4*H

// workspace layout in floats
#define REL_OFF 0
#define REL_SZ (VOCAB * HH)
#define A_OFF (REL_OFF + REL_SZ)
#define A_SZ (BB * LL * HH)
#define BT_OFF (A_OFF + A_SZ)

// ---------------------------------------------------------------------------
// K1: relrow[d][h] = sum_k table[d,k] * Wh_w[k,h], k in [0,1024)
// table[d,k] = sin/cos(d / 10000^((k&~1)/1024))
// ---------------------------------------------------------------------------
__global__ __launch_bounds__(256) void k_relrow(const float* __restrict__ Wh_w,
                                                float* __restrict__ relrow) {
  __shared__ float tab[D4H];
  const int d = blockIdx.x;   // 0..254
  const int h = threadIdx.x;  // 0..255
  const float p = (float)d;
  for (int k = h; k < D4H; k += 256) {
    float expnt = (float)(k & ~1) / (float)D4H;
    float div = powf(10000.0f, expnt);
    float ang = p / div;
    tab[k] = (k & 1) ? cosf(ang) : sinf(ang);
  }
  __syncthreads();
  float acc = 0.0f;
  for (int k = 0; k < D4H; ++k)
    acc = fmaf(tab[k], Wh_w[k * HH + h], acc);
  relrow[d * HH + h] = acc;
}

// ---------------------------------------------------------------------------
// K2: At[b,l,h] = hidden[b,l,:] @ (We+Wd) + Wh_b ;  Bt = hidden @ (Ws-Wd)
// Wh_w rows: [0,H)=We, [H,2H)=Ws, [2H,3H)=Wd, [3H,4H)=Wm
// ---------------------------------------------------------------------------
__global__ __launch_bounds__(256) void k_ab(const float* __restrict__ hs,
                                            const float* __restrict__ Wh_w,
                                            const float* __restrict__ Wh_b,
                                            float* __restrict__ At,
                                            float* __restrict__ Bt) {
  __shared__ float row[HH];
  const int bl = blockIdx.x;
  const int h = threadIdx.x;
  row[h] = hs[bl * HH + h];
  __syncthreads();
  float aa = Wh_b[h];
  float bb = 0.0f;
  for (int k = 0; k < HH; ++k) {
    float x = row[k];
    float we = Wh_w[k * HH + h];
    float ws = Wh_w[(HH + k) * HH + h];
    float wd = Wh_w[(2 * HH + k) * HH + h];
    aa = fmaf(x, we + wd, aa);
    bb = fmaf(x, ws - wd, bb);
  }
  At[bl * HH + h] = aa;
  Bt[bl * HH + h] = bb;
}

// ---------------------------------------------------------------------------
// K3: one workgroup per (b,i). 8 waves, each owns two 64x64 (j,h) regions of
// the 256x256 "pre" matrix. fp32 WMMA for the rank-H GEMM, fused tanh and a
// second WMMA chain reducing over h against zero-padded Wo (C=5 -> N=16).
// ---------------------------------------------------------------------------
__global__ __launch_bounds__(256) void k_main(
    const float* __restrict__ hs, const int* __restrict__ mask,
    const float* __restrict__ Wh_w, const float* __restrict__ Wo_w,
    const float* __restrict__ Wo_b, const float* __restrict__ relrow,
    const float* __restrict__ At, const float* __restrict__ Bt,
    float* __restrict__ out) {
  __shared__ float hi_s[HH];            // hidden[b,i,:]
  __shared__ float at_s[HH];            // At[b,i,:]
  __shared__ float wo_s[HH * NC];       // Wo_w
  __shared__ float wob_s[NC];
  __shared__ float out_w[8][LL * NC];   // per-wave partial logits (no atomics)
  __shared__ float tile_s[8][16 * 16];  // per-wave tanh tile scratch

  const int bi = blockIdx.x;
  const int b = bi >> 8;
  const int i = bi & 255;
  const int tid = threadIdx.x;

  hi_s[tid] = hs[(b * LL + i) * HH + tid];
  at_s[tid] = At[(b * LL + i) * HH + tid];
  for (int t = tid; t < HH * NC; t += 256) wo_s[t] = Wo_w[t];
  for (int t = tid; t < 8 * LL * NC; t += 256) (&out_w[0][0])[t] = 0.0f;
  if (tid < NC) wob_s[tid] = Wo_b[tid];
  __syncthreads();

  const int wave = tid >> 5;
  const int lane = tid & 31;
  const int half = lane >> 4;  // 0: K+0/1 ; 1: K+2/3
  const int ln = lane & 15;
  const int sel = half * 2;

  const float* __restrict__ hb = hs + b * LL * HH;       // hidden[b] (L x H)
  const float* __restrict__ Wm = Wh_w + 3 * HH * HH;     // (H x H)
  const float* __restrict__ Btb = Bt + b * LL * HH;      // (L x H)
  float* __restrict__ myt = tile_s[wave];
  float* __restrict__ myout = out_w[wave];

  for (int reg = wave; reg < 16; reg += 8) {
    const int jr = (reg >> 2) * 64;
    const int hr = (reg & 3) * 64;

    v8f acc[4][4];
#pragma unroll
    for (int mt = 0; mt < 4; ++mt)
#pragma unroll
      for (int nt = 0; nt < 4; ++nt)
        acc[mt][nt] = (v8f){0.f, 0.f, 0.f, 0.f, 0.f, 0.f, 0.f, 0.f};

    // ---- main GEMM: pre[j,h] += sum_k hb[j,k] * (hi[k]*Wm[k,h]) ----
    for (int K = 0; K < HH; K += 4) {
      v2f afr[4];
#pragma unroll
      for (int mt = 0; mt < 4; ++mt) {
        int j = jr + mt * 16 + ln;
        afr[mt] = *(const v2f*)(hb + j * HH + K + sel);  // A 16x4: lane layout
      }
      v2f hv = *(const v2f*)(hi_s + K + sel);  // hi[K+sel], hi[K+sel+1]
      v2f bfr[4];
#pragma unroll
      for (int nt = 0; nt < 4; ++nt) {
        int hcol = hr + nt * 16 + ln;
        float w0 = Wm[(K + sel) * HH + hcol];
        float w1 = Wm[(K + sel + 1) * HH + hcol];
        bfr[nt].x = w0 * hv.x;
        bfr[nt].y = w1 * hv.y;
      }
#pragma unroll
      for (int mt = 0; mt < 4; ++mt)
#pragma unroll
        for (int nt = 0; nt < 4; ++nt)
          acc[mt][nt] = __builtin_amdgcn_wmma_f32_16x16x4_f32(
              false, afr[mt], false, bfr[nt], (short)0, acc[mt][nt], false,
              false);
    }

    // ---- epilogue: add terms, tanh, second WMMA vs padded Wo ----
    v8f oacc[4];
#pragma unroll
    for (int mt = 0; mt < 4; ++mt)
      oacc[mt] = (v8f){0.f, 0.f, 0.f, 0.f, 0.f, 0.f, 0.f, 0.f};

#pragma unroll
    for (int mt = 0; mt < 4; ++mt) {
      for (int nt = 0; nt < 4; ++nt) {
        int hcol = hr + nt * 16 + ln;  // C layout: n = lane&15
        float ah = at_s[hcol];
#pragma unroll
        for (int rr = 0; rr < 8; ++rr) {
          int m = rr + half * 8;  // C layout: M = r (+8 for upper lanes)
          int j = jr + mt * 16 + m;
          int dist = j - i;
          dist = (dist < -127) ? -127 : (dist > 127 ? 127 : dist);
          dist += 127;
          float v = acc[mt][nt][rr] + ah + Btb[j * HH + hcol] +
                    relrow[dist * HH + hcol];
          myt[m * 16 + ln] = tanhf(v);
        }
        __builtin_amdgcn_wave_barrier();  // wave-synchronous LDS reuse
        // out[j,c] += tanh_tile (16x16, K=h) @ Wo_pad (16x16, N=c padded)
#pragma unroll
        for (int kk = 0; kk < 4; ++kk) {
          int kl = kk * 4 + sel;
          v2f a2 = *(const v2f*)(myt + ln * 16 + kl);
          v2f b2;
          b2.x = 0.0f;
          b2.y = 0.0f;
          if (ln < NC) {
            int hg = hr + nt * 16 + kl;
            b2.x = wo_s[hg * NC + ln];
            b2.y = wo_s[(hg + 1) * NC + ln];
          }
          oacc[mt] = __builtin_amdgcn_wmma_f32_16x16x4_f32(
              false, a2, false, b2, (short)0, oacc[mt], false, false);
        }
        __builtin_amdgcn_wave_barrier();
      }
      // park this j-tile's partial h-reduction in the wave-private buffer
      if (ln < NC) {
#pragma unroll
        for (int rr = 0; rr < 8; ++rr) {
          int j = jr + mt * 16 + rr + half * 8;
          myout[j * NC + ln] += oacc[mt][rr];
        }
      }
#pragma unroll
      for (int q = 0; q < 8; ++q) oacc[mt][q] = 0.0f;
    }
  }
  __syncthreads();

  // ---- deterministic cross-wave reduction + bias + tri + mask + store ----
  const int j = tid;
  const bool ok = (mask[b * LL + i] != 0) && (mask[b * LL + j] != 0);
#pragma unroll
  for (int c = 0; c < NC; ++c) {
    float v = 0.0f;
#pragma unroll
    for (int w = 0; w < 8; ++w) v += out_w[w][j * NC + c];
    v += wob_s[c];
    if (j < i) v -= 1e12f;
    if (!ok) v = -__builtin_inff();
    out[((b * NC + c) * LL + i) * LL + j] = v;
  }
}

// ---------------------------------------------------------------------------
extern "C" void kernel_launch(void* const* d_in, const int* in_sizes, int n_in,
                              void* d_out, int out_size, void* d_ws,
                              size_t ws_size, hipStream_t stream) {
  const float* hs = (const float*)d_in[0];    // [B,L,H]
  const int* mask = (const int*)d_in[1];      // [B,L]
  const float* Wh_w = (const float*)d_in[2];  // [4H,H]
  const float* Wh_b = (const float*)d_in[3];  // [H]
  const float* Wo_w = (const float*)d_in[4];  // [H,C]
  const float* Wo_b = (const float*)d_in[5];  // [C]
  float* out = (float*)d_out;                 // [B,C,L,L]
  float* ws = (float*)d_ws;

  float* relrow = ws + REL_OFF;  // [255, H]
  float* At = ws + A_OFF;        // [B*L, H]
  float* Bt = ws + BT_OFF;       // [B*L, H]

  k_relrow<<<VOCAB, 256, 0, stream>>>(Wh_w, relrow);
  k_ab<<<BB * LL, 256, 0, stream>>>(hs, Wh_w, Wh_b, At, Bt);
  k_main<<<BB * LL, 256, 0, stream>>>(hs, mask, Wh_w, Wo_w, Wo_b, relrow, At,
                                      Bt, out);
}